// InfoNCE_contrastiveLoss_56066503082787
// MI455X (gfx1250) — compile-verified
//
#include <hip/hip_runtime.h>
#include <math.h>

typedef __attribute__((ext_vector_type(2))) float v2f;
typedef __attribute__((ext_vector_type(8))) float v8f;

#define NROWS   8192
#define DIM     256
#define TEMPINV 10.0f
#define SHIFT   10.0f
#define LDS_STRIDE 260   // 256 + 4 pad: fragment ds_load_b64 hits all 64 banks

// ---------------------------------------------------------------------------
// Kernel 1: per-index inverse norms, diagonal similarity, zero accumulators.
// ---------------------------------------------------------------------------
__global__ __launch_bounds__(256) void prep_kernel(
    const float* __restrict__ im, const float* __restrict__ cap,
    float* __restrict__ invIm, float* __restrict__ invCap,
    float* __restrict__ diag, float* __restrict__ rowA, float* __restrict__ rowB)
{
    __shared__ float red[256];
    const int i = blockIdx.x;
    const int t = threadIdx.x;
    const float a = im[i * DIM + t];
    const float b = cap[i * DIM + t];
    float res0, res1, res2;
    red[t] = a * a; __syncthreads();
    for (int s = 128; s > 0; s >>= 1) { if (t < s) red[t] += red[t + s]; __syncthreads(); }
    res0 = red[0]; __syncthreads();
    red[t] = b * b; __syncthreads();
    for (int s = 128; s > 0; s >>= 1) { if (t < s) red[t] += red[t + s]; __syncthreads(); }
    res1 = red[0]; __syncthreads();
    red[t] = a * b; __syncthreads();
    for (int s = 128; s > 0; s >>= 1) { if (t < s) red[t] += red[t + s]; __syncthreads(); }
    res2 = red[0];

    if (t == 0) {
        float ia = rsqrtf(fmaxf(res0, 1e-24f));
        float ib = rsqrtf(fmaxf(res1, 1e-24f));
        invIm[i]  = ia;
        invCap[i] = ib;
        diag[i]   = res2 * ia * ib;
    }
    if (t == 1) rowA[i] = 0.0f;
    if (t == 2) rowB[i] = 0.0f;
}

// ---------------------------------------------------------------------------
// Kernel 2: fused GEMM + sum-of-exp, one orientation.
// Block = 256 threads = 8 waves; wave owns 16 rows; block covers 128 rows.
// B chunk staged in double-buffered LDS (coalesced b128 global loads issued
// one chunk ahead). Fragment reads software-pipelined in groups of 8 k-steps
// so ds_load latency is covered by the previous group's 8 WMMAs.
// ---------------------------------------------------------------------------
__global__ __launch_bounds__(256, 1) void clip_pass_kernel(
    const float* __restrict__ Arows,   // [8192 x 256] raw
    const float* __restrict__ Brows,   // [8192 x 256] raw
    const float* __restrict__ invA,
    const float* __restrict__ invB,
    float* __restrict__ rowSum)
{
    __shared__ float smem[2][16 * LDS_STRIDE];   // 2 x 16.6 KB

    const int tid    = threadIdx.x;
    const int lane   = tid & 31;
    const int wave   = tid >> 5;
    const int lane15 = lane & 15;
    const int hi     = lane >> 4;

    const int rowBase = (blockIdx.x * 8 + wave) * 16;
    const int colBase = blockIdx.y * 1024;       // 64 chunks of 16 columns

    // --- Preload all 64 A fragments (16 rows x K=256) into VGPRs ---
    const float* aRow = Arows + (size_t)(rowBase + lane15) * DIM + 2 * hi;
    v2f aReg[64];
#pragma unroll
    for (int k = 0; k < 64; ++k)
        aReg[k] = *(const v2f*)(aRow + 4 * k);

    float rni[8];
#pragma unroll
    for (int v = 0; v < 8; ++v)
        rni[v] = invA[rowBase + v + 8 * hi];

    float rsum[8];
#pragma unroll
    for (int v = 0; v < 8; ++v) rsum[v] = 0.0f;

    // --- Cooperative B staging: 16 rows x 256 floats = 1024 float4 / block ---
    float4 rbuf[4];
#pragma unroll
    for (int i = 0; i < 4; ++i) {
        int f  = tid + 256 * i;
        int r  = f >> 6;
        int c4 = f & 63;
        rbuf[i] = *(const float4*)(Brows + (size_t)(colBase + r) * DIM + c4 * 4);
    }
#pragma unroll
    for (int i = 0; i < 4; ++i) {
        int f  = tid + 256 * i;
        int r  = f >> 6;
        int c4 = f & 63;
        *(float4*)(&smem[0][r * LDS_STRIDE + c4 * 4]) = rbuf[i];
    }
    __syncthreads();

    for (int c = 0; c < 64; ++c) {
        // issue global loads for next chunk before compute (latency hiding)
        if (c < 63) {
#pragma unroll
            for (int i = 0; i < 4; ++i) {
                int f  = tid + 256 * i;
                int r  = f >> 6;
                int c4 = f & 63;
                rbuf[i] = *(const float4*)(Brows +
                          (size_t)(colBase + (c + 1) * 16 + r) * DIM + c4 * 4);
            }
        }

        const float cinv = invB[colBase + c * 16 + lane15] * TEMPINV;
        const float* sbuf = &smem[c & 1][lane15 * LDS_STRIDE + 2 * hi];

        // two independent accumulator chains; fragment reads pipelined in
        // groups of 8 (group g+1 loads issued before group g's WMMAs)
        v8f c0 = {};
        v8f c1 = {};
        v2f cur[8], nxt[8];
#pragma unroll
        for (int k = 0; k < 8; ++k)
            cur[k] = *(const v2f*)(sbuf + 4 * k);
#pragma unroll
        for (int g = 0; g < 8; ++g) {
            if (g < 7) {
#pragma unroll
                for (int k = 0; k < 8; ++k)
                    nxt[k] = *(const v2f*)(sbuf + 4 * (8 * (g + 1) + k));
            }
#pragma unroll
            for (int k = 0; k < 8; k += 2) {
                c0 = __builtin_amdgcn_wmma_f32_16x16x4_f32(
                         false, aReg[8 * g + k],     false, cur[k],     (short)0, c0, false, false);
                c1 = __builtin_amdgcn_wmma_f32_16x16x4_f32(
                         false, aReg[8 * g + k + 1], false, cur[k + 1], (short)0, c1, false, false);
            }
            if (g < 7) {
#pragma unroll
                for (int k = 0; k < 8; ++k)
                    cur[k] = nxt[k];
            }
        }

#pragma unroll
        for (int v = 0; v < 8; ++v) {
            float x = (c0[v] + c1[v]) * rni[v] * cinv - SHIFT;
            rsum[v] += __expf(x);
        }

        __syncthreads();
        if (c < 63) {
#pragma unroll
            for (int i = 0; i < 4; ++i) {
                int f  = tid + 256 * i;
                int r  = f >> 6;
                int c4 = f & 63;
                *(float4*)(&smem[(c + 1) & 1][r * LDS_STRIDE + c4 * 4]) = rbuf[i];
            }
        }
        __syncthreads();
    }

    // reduce across the 16 lanes of each half-wave
#pragma unroll
    for (int v = 0; v < 8; ++v) {
        float s = rsum[v];
        s += __shfl_xor(s, 1, 32);
        s += __shfl_xor(s, 2, 32);
        s += __shfl_xor(s, 4, 32);
        s += __shfl_xor(s, 8, 32);
        rsum[v] = s;
    }
    if (lane15 == 0) {
#pragma unroll
        for (int v = 0; v < 8; ++v)
            atomicAdd(&rowSum[rowBase + v + 8 * hi], rsum[v]);
    }
}

// ---------------------------------------------------------------------------
// Kernel 3: combine both orientations into the scalar loss.
// ---------------------------------------------------------------------------
__global__ __launch_bounds__(256) void finalize_kernel(
    const float* __restrict__ rowA, const float* __restrict__ rowB,
    const float* __restrict__ diag, float* __restrict__ out)
{
    __shared__ float red[256];
    const int t = threadIdx.x;
    float part = 0.0f;
    for (int i = t; i < NROWS; i += 256) {
        part += logf(rowA[i]) + logf(rowB[i])
              + 2.0f * SHIFT - 2.0f * TEMPINV * diag[i];
    }
    red[t] = part; __syncthreads();
    for (int s = 128; s > 0; s >>= 1) { if (t < s) red[t] += red[t + s]; __syncthreads(); }
    if (t == 0) out[0] = red[0] / (2.0f * NROWS);
}

// ---------------------------------------------------------------------------
extern "C" void kernel_launch(void* const* d_in, const int* in_sizes, int n_in,
                              void* d_out, int out_size, void* d_ws, size_t ws_size,
                              hipStream_t stream)
{
    const float* im  = (const float*)d_in[0];
    const float* cap = (const float*)d_in[1];
    float* out = (float*)d_out;

    float* ws     = (float*)d_ws;
    float* invIm  = ws;            // [8192]
    float* invCap = ws + 8192;     // [8192]
    float* diag   = ws + 16384;    // [8192]
    float* rowA   = ws + 24576;    // [8192]
    float* rowB   = ws + 32768;    // [8192]

    prep_kernel<<<NROWS, 256, 0, stream>>>(im, cap, invIm, invCap, diag, rowA, rowB);

    dim3 grid(64, 8);
    clip_pass_kernel<<<grid, 256, 0, stream>>>(im,  cap, invIm,  invCap, rowA);
    clip_pass_kernel<<<grid, 256, 0, stream>>>(cap, im,  invCap, invIm,  rowB);

    finalize_kernel<<<1, 256, 0, stream>>>(rowA, rowB, diag, out);
}